// TransR_22385369547467
// MI455X (gfx1250) — compile-verified
//
#include <hip/hip_runtime.h>
#include <math.h>

typedef __attribute__((ext_vector_type(2))) float v2f;
typedef __attribute__((ext_vector_type(8))) float v8f;

#define NUM_REL   500
#define BATCH_N   8192
#define ED        256
#define RD        128
#define GROUP_M   16
#define MAX_GROUPS 1024
#define DT_STRIDE (ED + 4)   // 260 words: 16B-aligned rows AND conflict-free (260%64==4)

// ws layout (int32 indices):
//  [0    .. 511 ]  counts
//  [512  .. 1023]  cursors
//  [1024 .. 1535]  offsets
//  [1536]          numGroups
//  [1600 .. 2623]  grp_rel
//  [2624 .. 3647]  grp_start
//  [3648 .. 4671]  grp_cnt
//  [4672 .. 12863] bucket (sample ids sorted by relation)
#define WS_COUNTS   0
#define WS_CURSORS  512
#define WS_OFFSETS  1024
#define WS_NGROUPS  1536
#define WS_GRPREL   1600
#define WS_GRPSTART 2624
#define WS_GRPCNT   3648
#define WS_BUCKET   4672

__global__ void k_zero_ws(int* __restrict__ ws) {
    int i = blockIdx.x * blockDim.x + threadIdx.x;
    if (i < 1024) ws[i] = 0;            // counts + cursors
    if (i == 0)   ws[WS_NGROUPS] = 0;
}

__global__ void k_hist(const int* __restrict__ relations, int* __restrict__ ws) {
    int i = blockIdx.x * blockDim.x + threadIdx.x;
    if (i < BATCH_N) atomicAdd(&ws[WS_COUNTS + relations[i]], 1);
}

// Parallel scan + group-record emission: one 512-thread block, Hillis-Steele
// over (sample counts, group counts) in LDS, then each relation writes its own
// group records. Replaces the previous 1-thread serial scan (which would have
// cost ~500 dependent global round-trips).
__global__ __launch_bounds__(512) void k_scan_groups(int* __restrict__ ws) {
    __shared__ int sc[512];   // inclusive scan of sample counts
    __shared__ int sg[512];   // inclusive scan of group counts
    const int t = threadIdx.x;
    const int n  = (t < NUM_REL) ? ws[WS_COUNTS + t] : 0;
    const int ng = (n + GROUP_M - 1) / GROUP_M;
    sc[t] = n; sg[t] = ng;
    __syncthreads();
    for (int d = 1; d < 512; d <<= 1) {
        int a = 0, b = 0;
        if (t >= d) { a = sc[t - d]; b = sg[t - d]; }
        __syncthreads();
        sc[t] += a; sg[t] += b;
        __syncthreads();
    }
    const int off   = sc[t] - n;    // exclusive
    const int gbase = sg[t] - ng;
    if (t < NUM_REL) {
        ws[WS_OFFSETS + t] = off;
        for (int s = 0, g = 0; s < n; s += GROUP_M, ++g) {
            const int idx = gbase + g;
            if (idx < MAX_GROUPS) {
                ws[WS_GRPREL   + idx] = t;
                ws[WS_GRPSTART + idx] = off + s;
                const int rem = n - s;
                ws[WS_GRPCNT   + idx] = rem < GROUP_M ? rem : GROUP_M;
            }
        }
    }
    if (t == 511) {
        int total = sg[511];
        ws[WS_NGROUPS] = total < MAX_GROUPS ? total : MAX_GROUPS;
    }
}

__global__ void k_scatter(const int* __restrict__ relations, int* __restrict__ ws) {
    int i = blockIdx.x * blockDim.x + threadIdx.x;
    if (i < BATCH_N) {
        int r = relations[i];
        int pos = ws[WS_OFFSETS + r] + atomicAdd(&ws[WS_CURSORS + r], 1);
        ws[WS_BUCKET + pos] = i;
    }
}

// One block (8 waves of 32) per group of <=16 samples sharing one relation.
// D tile (16x256) = h - t staged in LDS; each wave computes one 16x16 output
// tile of D @ M[:, cb:cb+16] via 64 x v_wmma_f32_16x16x4_f32, adds r, squares,
// then a row reduction produces the L2 norm per sample.
__global__ __launch_bounds__(256) void k_transr_wmma(
    const int*   __restrict__ heads,
    const int*   __restrict__ tails,
    const float* __restrict__ entity_emb,
    const float* __restrict__ relation_emb,
    const float* __restrict__ proj_mats,
    const int*   __restrict__ ws,
    float*       __restrict__ out)
{
    const int numGroups = ws[WS_NGROUPS];
    const int g = blockIdx.x;
    if (g >= numGroups) return;

    const int rel   = ws[WS_GRPREL   + g];
    const int start = ws[WS_GRPSTART + g];
    const int cnt   = ws[WS_GRPCNT   + g];
    const int* bucket = ws + WS_BUCKET;

    __shared__ float Dt[GROUP_M][DT_STRIDE];
    __shared__ float Sq[GROUP_M][RD];
    __shared__ float Rv[RD];
    __shared__ int   Smp[GROUP_M];

    const int tid = threadIdx.x;

    if (tid < RD) Rv[tid] = relation_emb[(size_t)rel * RD + tid];
    if (tid < GROUP_M) Smp[tid] = (tid < cnt) ? bucket[start + tid] : -1;

    // Stage D = h - t with float4 traffic (rows >= cnt zero-padded: full EXEC WMMA).
    {
        const int row  = tid >> 4;          // 0..15
        const int cseg = (tid & 15) * 16;   // 16 floats (4 float4) per thread
        const int smp = (row < cnt) ? bucket[start + row] : -1;
        float4* dst = reinterpret_cast<float4*>(&Dt[row][cseg]);
        if (smp >= 0) {
            const float4* hp = reinterpret_cast<const float4*>(entity_emb + (size_t)heads[smp] * ED + cseg);
            const float4* tp = reinterpret_cast<const float4*>(entity_emb + (size_t)tails[smp] * ED + cseg);
            #pragma unroll
            for (int j = 0; j < 4; ++j) {
                float4 hv = hp[j], tv = tp[j];
                dst[j] = make_float4(hv.x - tv.x, hv.y - tv.y, hv.z - tv.z, hv.w - tv.w);
            }
        } else {
            #pragma unroll
            for (int j = 0; j < 4; ++j)
                dst[j] = make_float4(0.f, 0.f, 0.f, 0.f);
        }
    }
    __syncthreads();

    const int wave = tid >> 5;
    const int lane = tid & 31;
    const int hl   = lane & 15;
    const int koff = (lane < 16) ? 0 : 2;   // A/B frag: lanes 0-15 hold K0/K1, lanes 16-31 K2/K3
    const int cb   = wave * 16;             // this wave's 16-column slice of RD
    const float* M = proj_mats + (size_t)rel * ED * RD;

    v8f acc = {};
    #pragma unroll 8
    for (int k0 = 0; k0 < ED; k0 += 4) {
        v2f a, b;
        a.x = Dt[hl][k0 + koff];
        a.y = Dt[hl][k0 + koff + 1];
        const float* mp = M + (size_t)(k0 + koff) * RD + cb + hl;
        b.x = mp[0];
        b.y = mp[RD];
        acc = __builtin_amdgcn_wmma_f32_16x16x4_f32(
            /*neg_a=*/false, a, /*neg_b=*/false, b,
            /*c_mod=*/(short)0, acc, /*reuse_a=*/false, /*reuse_b=*/false);
    }

    // diff = proj + r ; store squares (C/D layout: VGPR v -> row v (lanes<16) / v+8)
    const float rv = Rv[cb + hl];
    #pragma unroll
    for (int v = 0; v < 8; ++v) {
        const int row = v + ((lane < 16) ? 0 : 8);
        const float val = acc[v] + rv;
        Sq[row][cb + hl] = val * val;
    }
    __syncthreads();

    if (tid < GROUP_M && tid < cnt) {
        float s = 0.0f;
        #pragma unroll 8
        for (int c = 0; c < RD; ++c) s += Sq[tid][c];
        out[Smp[tid]] = sqrtf(s);
    }
}

extern "C" void kernel_launch(void* const* d_in, const int* in_sizes, int n_in,
                              void* d_out, int out_size, void* d_ws, size_t ws_size,
                              hipStream_t stream) {
    const int*   heads        = (const int*)  d_in[0];
    const int*   relations    = (const int*)  d_in[1];
    const int*   tails        = (const int*)  d_in[2];
    const float* entity_emb   = (const float*)d_in[3];
    const float* relation_emb = (const float*)d_in[4];
    const float* proj_mats    = (const float*)d_in[5];
    float* out = (float*)d_out;
    int*   ws  = (int*)d_ws;

    k_zero_ws    <<<4, 256, 0, stream>>>(ws);
    k_hist       <<<BATCH_N / 256, 256, 0, stream>>>(relations, ws);
    k_scan_groups<<<1, 512, 0, stream>>>(ws);
    k_scatter    <<<BATCH_N / 256, 256, 0, stream>>>(relations, ws);
    k_transr_wmma<<<MAX_GROUPS, 256, 0, stream>>>(heads, tails, entity_emb,
                                                  relation_emb, proj_mats, ws, out);
}